// GraphAttentionLayer_80977313398923
// MI455X (gfx1250) — compile-verified
//
#include <hip/hip_runtime.h>
#include <hip/hip_bf16.h>
#include <stdint.h>

typedef float v2f __attribute__((ext_vector_type(2)));
typedef float v8f __attribute__((ext_vector_type(8)));

#define N_NODES 4096
#define F_IN    512
#define F_OUT   256
#define GAT_ALPHA 0.2f
#define GAT_NEG_INF -9.0e15f

// ---------------------------------------------------------------------------
// Kernel 1: wh = h @ W + b   (4096x512 @ 512x256, fp32 WMMA 16x16x4)
// One wave per 16x16 output tile. 4 waves / block -> 1024 blocks.
// ---------------------------------------------------------------------------
__global__ __launch_bounds__(128) void gat_proj_kernel(
    const float* __restrict__ h, const float* __restrict__ W,
    const float* __restrict__ bias, float* __restrict__ wh)
{
    const int lane = threadIdx.x & 31;
    const int wave = threadIdx.x >> 5;                // 0..3
    const int tile = blockIdx.x * 4 + wave;           // 0..4095
    const int rt = tile >> 4;                         // row tile 0..255
    const int ct = tile & 15;                         // col tile 0..15
    const int rowbase = rt * 16;
    const int colbase = ct * 16;
    const int lrow = lane & 15;
    const int lhi  = lane >> 4;                       // 0 or 1

    // C/D layout: acc[r] @ lane -> (row = rowbase + r + 8*lhi, col = colbase + lrow)
    v8f acc;
    const float bv = bias[colbase + lrow];
#pragma unroll
    for (int r = 0; r < 8; ++r) acc[r] = bv;

    const float* hrow = h + (size_t)(rowbase + lrow) * F_IN;

    for (int k = 0; k < F_IN; k += 4) {
        const int ka = k + 2 * lhi;                   // this lane's K pair
        v2f a, b;
        // A layout (16x4 f32): lane<16 holds K={k,k+1}, lane>=16 holds K={k+2,k+3}
        a.x = hrow[ka];
        a.y = hrow[ka + 1];
        // B layout (4x16 f32): VGPR0 = rows {k, k+2}, VGPR1 = rows {k+1, k+3}
        b.x = W[(size_t)ka * F_OUT + colbase + lrow];
        b.y = W[(size_t)(ka + 1) * F_OUT + colbase + lrow];
        acc = __builtin_amdgcn_wmma_f32_16x16x4_f32(
            /*neg_a=*/false, a, /*neg_b=*/false, b,
            /*c_mod=*/(short)0, acc, /*reuse_a=*/false, /*reuse_b=*/false);
    }

#pragma unroll
    for (int r = 0; r < 8; ++r)
        wh[(size_t)(rowbase + r + 8 * lhi) * F_OUT + colbase + lrow] = acc[r];
}

// ---------------------------------------------------------------------------
// Kernel 2: f_src[i] = wh[i,:].a1 + a_b ; f_dst[i] = wh[i,:].a2
// One wave (32 lanes) per row, shuffle reduction. 8 waves/block -> 512 blocks.
// ---------------------------------------------------------------------------
__global__ __launch_bounds__(256) void gat_attvec_kernel(
    const float* __restrict__ wh, const float* __restrict__ a1,
    const float* __restrict__ a2, const float* __restrict__ a_b,
    float* __restrict__ fsrc, float* __restrict__ fdst)
{
    const int lane = threadIdx.x & 31;
    const int wave = threadIdx.x >> 5;
    const int row  = blockIdx.x * 8 + wave;
    const float* whr = wh + (size_t)row * F_OUT;

    float s1 = 0.0f, s2 = 0.0f;
#pragma unroll
    for (int i = lane; i < F_OUT; i += 32) {
        const float v = whr[i];
        s1 += v * a1[i];
        s2 += v * a2[i];
    }
#pragma unroll
    for (int off = 16; off > 0; off >>= 1) {
        s1 += __shfl_xor(s1, off, 32);
        s2 += __shfl_xor(s2, off, 32);
    }
    if (lane == 0) {
        fsrc[row] = s1 + a_b[0];   // fold bias of proj_a into the src half
        fdst[row] = s2;
    }
}

// ---------------------------------------------------------------------------
// Kernel 3: fused masked softmax (online, flash-style) + att@wh + ELU.
// Block = 256 threads (8 waves) owns 16 output rows x 256 cols.
// Wave w owns cols [32w, 32w+32): two 16x16 fp32 WMMA accumulators.
// Per 16-wide column tile:
//   - issue GLOBAL_LOAD_ASYNC_TO_LDS_B128 to stage the 16x256 wh tile into
//     LDS (ASYNCcnt), overlapping the DMA with score-tile construction
//   - 256 threads cooperatively build masked leaky-relu score tile in LDS
//   - threads 0..15 update per-row running max/sum, emit P and scale to LDS
//   - s_wait_asynccnt 0 + barrier, then every wave rescales its acc, reads P
//     in A-layout and the wh tile in B-layout from LDS (conflict-free
//     padding), and issues 8 x v_wmma_f32_16x16x4_f32.
// ---------------------------------------------------------------------------
#define WH_PAD 264   // 256 + 8: 2-row delta = 16 banks mod 64 -> no conflicts

__global__ __launch_bounds__(256) void gat_softmax_matmul_kernel(
    const int* __restrict__ adj, const float* __restrict__ wh,
    const float* __restrict__ fsrc, const float* __restrict__ fdst,
    float* __restrict__ out)
{
    __shared__ float sWh[16][WH_PAD]; // async-staged wh j-tile
    __shared__ float sE[16][17];      // scores (padded: no LDS bank conflicts)
    __shared__ float sP[16][17];      // exp(e - m_new)
    __shared__ float sM[16];          // running row max
    __shared__ float sL[16];          // running row sum
    __shared__ float sScale[16];      // exp(m_old - m_new) for acc rescale

    const int tid  = threadIdx.x;    // 0..255
    const int lane = tid & 31;
    const int wave = tid >> 5;       // 0..7
    const int rowbase = blockIdx.x * 16;

    const int erow = tid >> 4;       // 0..15 (score-tile row this thread fills)
    const int ecol = tid & 15;       // 0..15 (consecutive -> coalesced adj)

    const int lrow = lane & 15;
    const int lhi  = lane >> 4;

    if (tid < 16) { sM[tid] = -__builtin_inff(); sL[tid] = 0.0f; }
    __syncthreads();

    const float fs_r = fsrc[rowbase + erow];
    const int*  adjrow = adj + (size_t)(rowbase + erow) * N_NODES;

    // async-staging assignment: thread t covers sWh[t>>4][(t&15)*16 .. +15]
    const int srow = tid >> 4;
    const int scol = (tid & 15) * 16;
    const uint32_t lds_base = (uint32_t)(uintptr_t)(&sWh[srow][scol]);

    const int col0 = wave * 32;      // wave's first N-tile
    const int col1 = wave * 32 + 16; // wave's second N-tile
    v8f acc0 = {}; v8f acc1 = {};

    for (int jt = 0; jt < N_NODES / 16; ++jt) {
        const int jbase = jt * 16;

        // 0. kick off async DMA of this j-tile of wh into LDS (ASYNCcnt)
        {
            const float* gsrc = wh + (size_t)(jbase + srow) * F_OUT + scol;
#pragma unroll
            for (int c = 0; c < 4; ++c) {
                asm volatile("global_load_async_to_lds_b128 %0, %1, off"
                             :: "v"(lds_base + c * 16), "v"(gsrc + c * 4)
                             : "memory");
            }
        }

        // prefetch next adj tile (global_prefetch_b8)
        if (jt + 1 < N_NODES / 16)
            __builtin_prefetch(&adjrow[jbase + 16 + ecol], 0, 3);

        // 1. masked leaky-relu score tile
        const int a = adjrow[jbase + ecol];
        float ev;
        if (a > 0) {
            const float x = fs_r + fdst[jbase + ecol];
            ev = (x > 0.0f) ? x : GAT_ALPHA * x;
        } else {
            ev = GAT_NEG_INF;
        }
        sE[erow][ecol] = ev;

        // every wave drains its own async transfers, then block-barrier:
        // after this barrier both sE and the full sWh tile are visible.
        asm volatile("s_wait_asynccnt 0x0" ::: "memory");
        __syncthreads();

        // 2. per-row online softmax stats (threads 0..15, one row each)
        if (tid < 16) {
            const float m_old = sM[tid];
            float tmax = sE[tid][0];
#pragma unroll
            for (int c = 1; c < 16; ++c) tmax = fmaxf(tmax, sE[tid][c]);
            const float m_new = fmaxf(m_old, tmax);
            const float sc = __expf(m_old - m_new);   // first iter: exp(-inf)=0
            float psum = 0.0f;
#pragma unroll
            for (int c = 0; c < 16; ++c) {
                const float p = __expf(sE[tid][c] - m_new);
                sP[tid][c] = p;
                psum += p;
            }
            sM[tid] = m_new;
            sScale[tid] = sc;
            sL[tid] = sL[tid] * sc + psum;
        }
        __syncthreads();

        // 3. rescale accumulators (row of acc[r] = r + 8*lhi)
#pragma unroll
        for (int r = 0; r < 8; ++r) {
            const float s = sScale[r + 8 * lhi];
            acc0[r] *= s;
            acc1[r] *= s;
        }

        // 4. acc += P(16x16) @ wh_tile(16x32) via 4 K-chunks of fp32 WMMA,
        //    A from sP, B from the async-staged sWh
#pragma unroll
        for (int kc = 0; kc < 4; ++kc) {
            const int k0 = kc * 4 + 2 * lhi;          // lane's K pair
            v2f av;
            av.x = sP[lrow][k0];
            av.y = sP[lrow][k0 + 1];
            v2f b0, b1;
            b0.x = sWh[k0][col0 + lrow];  b0.y = sWh[k0 + 1][col0 + lrow];
            b1.x = sWh[k0][col1 + lrow];  b1.y = sWh[k0 + 1][col1 + lrow];
            acc0 = __builtin_amdgcn_wmma_f32_16x16x4_f32(
                false, av, false, b0, (short)0, acc0, false, false);
            acc1 = __builtin_amdgcn_wmma_f32_16x16x4_f32(
                false, av, false, b1, (short)0, acc1, false, false);
        }
        __syncthreads();   // protect sE/sP/sWh before next tile overwrites them
    }

    // 5. normalize by row sum, ELU, store
#pragma unroll
    for (int r = 0; r < 8; ++r) {
        const int row = rowbase + r + 8 * lhi;
        const float inv = 1.0f / sL[r + 8 * lhi];
        float v0 = acc0[r] * inv;
        float v1 = acc1[r] * inv;
        v0 = (v0 > 0.0f) ? v0 : expm1f(v0);
        v1 = (v1 > 0.0f) ? v1 : expm1f(v1);
        out[(size_t)row * F_OUT + col0 + lrow] = v0;
        out[(size_t)row * F_OUT + col1 + lrow] = v1;
    }
}

// ---------------------------------------------------------------------------
// Launch: inputs in setup_inputs() order: adj, h, W, b, a1, a2, a_b
// ws layout: wh (4096*256 f32) | fsrc (4096 f32) | fdst (4096 f32)
// ---------------------------------------------------------------------------
extern "C" void kernel_launch(void* const* d_in, const int* in_sizes, int n_in,
                              void* d_out, int out_size, void* d_ws, size_t ws_size,
                              hipStream_t stream) {
    (void)in_sizes; (void)n_in; (void)out_size; (void)ws_size;
    const int*   adj = (const int*)d_in[0];
    const float* h   = (const float*)d_in[1];
    const float* W   = (const float*)d_in[2];
    const float* b   = (const float*)d_in[3];
    const float* a1  = (const float*)d_in[4];
    const float* a2  = (const float*)d_in[5];
    const float* a_b = (const float*)d_in[6];
    float* out = (float*)d_out;

    float* wh   = (float*)d_ws;
    float* fsrc = wh + (size_t)N_NODES * F_OUT;
    float* fdst = fsrc + N_NODES;

    // 4096 tiles of wh / 4 waves per block
    gat_proj_kernel<<<1024, 128, 0, stream>>>(h, W, b, wh);
    // 4096 rows / 8 waves per block
    gat_attvec_kernel<<<512, 256, 0, stream>>>(wh, a1, a2, a_b, fsrc, fdst);
    // 256 row-blocks of 16
    gat_softmax_matmul_kernel<<<256, 256, 0, stream>>>(adj, wh, fsrc, fdst, out);
}